// Mamba2Block_25280177504375
// MI455X (gfx1250) — compile-verified
//
#include <hip/hip_runtime.h>
#include <hip/hip_bf16.h>

// ---------------- problem constants ----------------
#define BATCH     4
#define SEQLEN    2048
#define D_MODEL   1024
#define D_STATE   64
#define N_HEADS   32
#define D_INNER   2048
#define HEAD_DIM  64
#define KSIZE     4
#define IN_PROJ   8224          // 2*D_INNER + 2*N_HEADS*D_STATE + N_HEADS
#define ROWS      (BATCH * SEQLEN)   // 8192
// proj column offsets
#define OFF_X     0
#define OFF_Z     2048
#define OFF_B     4096
#define OFF_C     6144
#define OFF_DT    8192

typedef __attribute__((ext_vector_type(2))) float v2f;
typedef __attribute__((ext_vector_type(8))) float v8f;

// ======================================================================
// Kernel 1: LayerNorm over D_MODEL=1024, one 256-thread block per row.
// ======================================================================
__global__ __launch_bounds__(256) void ln_kernel(const float* __restrict__ x,
                                                 const float* __restrict__ w,
                                                 const float* __restrict__ b,
                                                 float* __restrict__ xn) {
    const int row  = blockIdx.x;
    const int tid  = threadIdx.x;
    const int lane = tid & 31, wave = tid >> 5;
    const float* xr = x + (size_t)row * D_MODEL;
    float4 v = reinterpret_cast<const float4*>(xr)[tid];

    __shared__ float sm[8];
    float s = v.x + v.y + v.z + v.w;
    for (int off = 16; off > 0; off >>= 1) s += __shfl_xor(s, off, 32);
    if (lane == 0) sm[wave] = s;
    __syncthreads();
    float mean = 0.f;
    for (int i = 0; i < 8; ++i) mean += sm[i];
    mean *= (1.0f / (float)D_MODEL);
    __syncthreads();

    float dx = v.x - mean, dy = v.y - mean, dz = v.z - mean, dw = v.w - mean;
    float s2 = dx * dx + dy * dy + dz * dz + dw * dw;
    for (int off = 16; off > 0; off >>= 1) s2 += __shfl_xor(s2, off, 32);
    if (lane == 0) sm[wave] = s2;
    __syncthreads();
    float var = 0.f;
    for (int i = 0; i < 8; ++i) var += sm[i];
    var *= (1.0f / (float)D_MODEL);
    const float rstd = rsqrtf(var + 1e-5f);

    float4 wv = reinterpret_cast<const float4*>(w)[tid];
    float4 bv = reinterpret_cast<const float4*>(b)[tid];
    float4 o;
    o.x = dx * rstd * wv.x + bv.x;
    o.y = dy * rstd * wv.y + bv.y;
    o.z = dz * rstd * wv.z + bv.z;
    o.w = dw * rstd * wv.w + bv.w;
    reinterpret_cast<float4*>(xn + (size_t)row * D_MODEL)[tid] = o;
}

// ======================================================================
// Kernel 2: NT GEMM, fp32 via V_WMMA_F32_16X16X4_F32.
//   C[m,n] = sum_k A[m,k] * W[n,k]  (+ optional residual[m,n])
// Block: 256 threads (8 wave32), 128(M) x 128(N) tile, K staged by 16.
// Wave grid 4(M) x 2(N); wave tile 32x64 = 2 M-tiles x 4 N-tiles
// (8 v8f accumulators) so each B fragment feeds 2 WMMAs and each A
// fragment feeds 4 -> halves LDS reads per WMMA vs a 16x64 wave tile.
// ======================================================================
__global__ __launch_bounds__(256) void gemm_nt_wmma(const float* __restrict__ A,
                                                    const float* __restrict__ W,
                                                    float* __restrict__ C,
                                                    const float* __restrict__ resid,
                                                    int M, int N, int K, int ldc) {
    __shared__ float As[128][17];  // padded: conflict-free M-major fragment reads
    __shared__ float Bs[16][129];  // padded: conflict-free transposed fill

    const int m0 = blockIdx.y * 128;
    const int n0 = blockIdx.x * 128;
    const int tid  = threadIdx.x;
    const int lane = tid & 31;
    const int wave = tid >> 5;
    const int wy = wave & 3;       // M sub-tile (32 rows each)
    const int wx = wave >> 2;      // N half (64 cols)

    v8f acc[2][4];
    const v8f vzero = {0.f, 0.f, 0.f, 0.f, 0.f, 0.f, 0.f, 0.f};
    for (int i = 0; i < 2; ++i)
        for (int t = 0; t < 4; ++t) acc[i][t] = vzero;

    // A loader: 128x16 floats; thread -> (row tid>>1, 8 cols). B loader: (col tid>>1, 8 k).
    const int ar = tid >> 1, ac = (tid & 1) * 8;
    const int bn = tid >> 1, bk = (tid & 1) * 8;

    for (int kk = 0; kk < K; kk += 16) {
        // ---- stage A 128x16 ----
        {
            const float* src = A + (size_t)(m0 + ar) * K + kk + ac;
            float4 v0 = reinterpret_cast<const float4*>(src)[0];
            float4 v1 = reinterpret_cast<const float4*>(src)[1];
            As[ar][ac + 0] = v0.x; As[ar][ac + 1] = v0.y;
            As[ar][ac + 2] = v0.z; As[ar][ac + 3] = v0.w;
            As[ar][ac + 4] = v1.x; As[ar][ac + 5] = v1.y;
            As[ar][ac + 6] = v1.z; As[ar][ac + 7] = v1.w;
        }
        // ---- stage B 16x128 (Bs[k][n] = W[n0+n][kk+k]), guarded on N ----
        {
            float4 v0, v1;
            if (n0 + bn < N) {
                const float* src = W + (size_t)(n0 + bn) * K + kk + bk;
                v0 = reinterpret_cast<const float4*>(src)[0];
                v1 = reinterpret_cast<const float4*>(src)[1];
            } else {
                v0 = make_float4(0.f, 0.f, 0.f, 0.f);
                v1 = v0;
            }
            Bs[bk + 0][bn] = v0.x; Bs[bk + 1][bn] = v0.y;
            Bs[bk + 2][bn] = v0.z; Bs[bk + 3][bn] = v0.w;
            Bs[bk + 4][bn] = v1.x; Bs[bk + 5][bn] = v1.y;
            Bs[bk + 6][bn] = v1.z; Bs[bk + 7][bn] = v1.w;
        }
        // prefetch next K tile while this one is consumed
        if (kk + 16 < K) {
            __builtin_prefetch(A + (size_t)(m0 + ar) * K + kk + 16 + ac, 0, 1);
            if (n0 + bn < N)
                __builtin_prefetch(W + (size_t)(n0 + bn) * K + kk + 16 + bk, 0, 1);
        }
        __syncthreads();

        // ---- 4 WMMA K-steps of 4 over the staged 16; 32 WMMAs per wave ----
        const int mrow = lane & 15;
        const int ksel = (lane >> 4) << 1;   // lanes 0-15 -> K{0,1}; 16-31 -> K{2,3}
        #pragma unroll
        for (int k4 = 0; k4 < 4; ++k4) {
            const int kb = k4 * 4 + ksel;
            v2f af[2];
            #pragma unroll
            for (int i = 0; i < 2; ++i) {
                af[i].x = As[wy * 32 + i * 16 + mrow][kb];
                af[i].y = As[wy * 32 + i * 16 + mrow][kb + 1];
            }
            #pragma unroll
            for (int t = 0; t < 4; ++t) {
                const int n = wx * 64 + t * 16 + mrow;
                v2f bf;
                bf.x = Bs[kb][n];
                bf.y = Bs[kb + 1][n];
                #pragma unroll
                for (int i = 0; i < 2; ++i) {
                    acc[i][t] = __builtin_amdgcn_wmma_f32_16x16x4_f32(
                        false, af[i], false, bf, (short)0, acc[i][t], false, false);
                }
            }
        }
        __syncthreads();
    }

    // ---- epilogue: C layout VGPR r / lane: (lane<16 ? M=r : M=r+8, N=lane&15) ----
    const int colL = lane & 15;
    const int rowB = (lane >> 4) * 8;
    for (int i = 0; i < 2; ++i) {
        for (int t = 0; t < 4; ++t) {
            const int n = n0 + wx * 64 + t * 16 + colL;
            if (n < N) {
                #pragma unroll
                for (int r = 0; r < 8; ++r) {
                    const int m = m0 + wy * 32 + i * 16 + rowB + r;
                    float v = acc[i][t][r];
                    if (resid) v += resid[(size_t)m * ldc + n];
                    C[(size_t)m * (size_t)ldc + n] = v;
                }
            }
        }
    }
}

// ======================================================================
// Kernel 3: causal depthwise conv (K=4) + SiLU on proj[:, 0:2048] -> xbr
// ======================================================================
__global__ __launch_bounds__(256) void conv_silu_kernel(const float* __restrict__ proj,
                                                        const float* __restrict__ cw,
                                                        const float* __restrict__ cb,
                                                        float* __restrict__ xbr) {
    const size_t row = blockIdx.x;          // b*SEQLEN + l
    const int l = (int)(row & (SEQLEN - 1));
    #pragma unroll
    for (int j = 0; j < 8; ++j) {
        const int d = threadIdx.x + j * 256;
        float acc = cb[d];
        #pragma unroll
        for (int k = 0; k < KSIZE; ++k) {
            const int ls = l - (KSIZE - 1) + k;
            if (ls >= 0)
                acc += cw[d * KSIZE + k] *
                       proj[(row - (KSIZE - 1) + k) * IN_PROJ + d];
        }
        const float s = acc / (1.0f + __expf(-acc));   // SiLU
        xbr[row * D_INNER + d] = s;
    }
}

// ======================================================================
// Kernel 4: dt = softplus(dt) * (1 - weekend_mask), in place in proj
// ======================================================================
__global__ __launch_bounds__(256) void dt_kernel(float* __restrict__ proj,
                                                 const float* __restrict__ mask) {
    const int idx = blockIdx.x * 256 + threadIdx.x;
    if (idx >= ROWS * N_HEADS) return;
    const int row = idx >> 5;
    const int h   = idx & 31;
    float* p = proj + (size_t)row * IN_PROJ + OFF_DT + h;
    const float v  = *p;
    const float sp = (v > 20.f) ? v : log1pf(__expf(v));
    *p = sp * (1.0f - mask[row]);
}

// ======================================================================
// Kernel 5: selective scan, one block per (batch, head).
// State [N_STATE=64 x P=64] in registers: thread (g = tid>>6, p = tid&63)
// owns h[n = g*16..g*16+15][p]. Shared exp(A*dt) / dt*B computed once by
// 64 threads per step (LDS broadcast reads). Gated output written in
// place over consumed x_branch.
// ======================================================================
__global__ __launch_bounds__(256) void scan_kernel(const float* __restrict__ proj,
                                                   float* __restrict__ xy,   // in: x_branch, out: y*silu(z)
                                                   const float* __restrict__ A_log) {
    const int b = blockIdx.x >> 5;
    const int h = blockIdx.x & 31;
    const int tid = threadIdx.x;
    const int g = tid >> 6;
    const int p = tid & 63;

    __shared__ float sab[64], sdB[64], sC[64], sx[64], sz[64];
    __shared__ float red[256];

    float aN = 0.f;
    if (tid < 64) aN = -__expf(A_log[h * D_STATE + tid]);

    float hst[16];
    #pragma unroll
    for (int i = 0; i < 16; ++i) hst[i] = 0.f;

    const size_t rowBase = (size_t)b * SEQLEN;
    for (int l = 0; l < SEQLEN; ++l) {
        const size_t row = rowBase + l;
        const float* pr = proj + row * IN_PROJ;
        if (tid < 64) {
            const float dtv = pr[OFF_DT + h];                 // broadcast via cache
            const float Bv  = pr[OFF_B + h * D_STATE + tid];
            sdB[tid] = dtv * Bv;
            sab[tid] = __expf(aN * dtv);
        } else if (tid < 128) {
            sC[tid - 64] = pr[OFF_C + h * D_STATE + (tid - 64)];
        } else if (tid < 192) {
            sx[tid - 128] = xy[row * D_INNER + h * HEAD_DIM + (tid - 128)];
        } else {
            sz[tid - 192] = pr[OFF_Z + h * HEAD_DIM + (tid - 192)];
        }
        __syncthreads();

        const float xp = sx[p];
        const float zv = sz[p];
        float acc = 0.f;
        #pragma unroll
        for (int i = 0; i < 16; ++i) {
            const int n = g * 16 + i;                          // same addr across wave -> LDS bcast
            hst[i] = sab[n] * hst[i] + sdB[n] * xp;
            acc += sC[n] * hst[i];
        }
        red[tid] = acc;
        __syncthreads();

        if (tid < 64) {
            const float y = red[p] + red[64 + p] + red[128 + p] + red[192 + p];
            const float gate = zv / (1.0f + __expf(-zv));      // silu(z)
            xy[row * D_INNER + h * HEAD_DIM + p] = y * gate;
        }
        // red reads finish before next iteration's post-barrier writes; safe.
    }
}

// ======================================================================
// Host-side launch
// ======================================================================
extern "C" void kernel_launch(void* const* d_in, const int* in_sizes, int n_in,
                              void* d_out, int out_size, void* d_ws, size_t ws_size,
                              hipStream_t stream) {
    const float* x        = (const float*)d_in[0];
    const float* mask     = (const float*)d_in[1];
    const float* norm_w   = (const float*)d_in[2];
    const float* norm_b   = (const float*)d_in[3];
    const float* in_proj  = (const float*)d_in[4];
    const float* conv_w   = (const float*)d_in[5];
    const float* conv_b   = (const float*)d_in[6];
    const float* A_log    = (const float*)d_in[7];
    const float* out_proj = (const float*)d_in[8];
    float* out = (float*)d_out;

    char* ws = (char*)d_ws;
    const size_t XN_BYTES   = (size_t)ROWS * D_MODEL * sizeof(float);   // 32 MB
    const size_t PROJ_BYTES = (size_t)ROWS * IN_PROJ * sizeof(float);   // 269 MB
    float* xn   = (float*)(ws);
    float* proj = (float*)(ws + XN_BYTES);
    float* xbr  = (float*)(ws + XN_BYTES + PROJ_BYTES);                 // also gated y

    // 1) LayerNorm
    ln_kernel<<<ROWS, 256, 0, stream>>>(x, norm_w, norm_b, xn);

    // 2) proj = xn @ in_proj_w^T   [8192,1024] x [8224,1024]^T
    gemm_nt_wmma<<<dim3((IN_PROJ + 127) / 128, ROWS / 128), 256, 0, stream>>>(
        xn, in_proj, proj, nullptr, ROWS, IN_PROJ, D_MODEL, IN_PROJ);

    // 3) causal depthwise conv + SiLU -> x_branch
    conv_silu_kernel<<<ROWS, 256, 0, stream>>>(proj, conv_w, conv_b, xbr);

    // 4) dt = softplus(dt) * (1 - mask), in place
    dt_kernel<<<(ROWS * N_HEADS + 255) / 256, 256, 0, stream>>>(proj, mask);

    // 5) selective scan + SiLU(z) gating (writes over xbr)
    scan_kernel<<<BATCH * N_HEADS, 256, 0, stream>>>(proj, xbr, A_log);

    // 6) out = y @ out_proj_w^T + residual   [8192,2048] x [1024,2048]^T
    gemm_nt_wmma<<<dim3(D_MODEL / 128, ROWS / 128), 256, 0, stream>>>(
        xbr, out_proj, out, x, ROWS, D_MODEL, D_INNER, D_MODEL);
}